// QM9_GCN_62045097558428
// MI455X (gfx1250) — compile-verified
//
#include <hip/hip_runtime.h>
#include <math.h>

typedef __attribute__((ext_vector_type(2))) float v2f;
typedef __attribute__((ext_vector_type(8))) float v8f;

#define FIN 11
#define HID 32
#define NCLS 16

// ---------------- utility kernels ----------------

__global__ void zero_kernel(float* __restrict__ p, int n) {
    int i = blockIdx.x * blockDim.x + threadIdx.x;
    if (i < n) p[i] = 0.0f;
}

__global__ void deg_kernel(const int* __restrict__ dst, float* __restrict__ deg, int E) {
    int e = blockIdx.x * blockDim.x + threadIdx.x;
    if (e < E) unsafeAtomicAdd(&deg[dst[e]], 1.0f);
}

__global__ void dinv_kernel(float* __restrict__ dinv, int n) {
    int i = blockIdx.x * blockDim.x + threadIdx.x;
    if (i < n) dinv[i] = rsqrtf(dinv[i] + 1.0f);
}

// ---------------- layer-1 GEMM (K=11, scalar: trivial FLOPs) ----------------

__global__ void gemm_in_kernel(const float* __restrict__ x, const float* __restrict__ W,
                               float* __restrict__ h, int n) {
    int tid = blockIdx.x * blockDim.x + threadIdx.x;
    if (tid >= n * HID) return;
    int node = tid / HID;
    int c    = tid % HID;
    const float* xr = x + (long long)node * FIN;
    float s = 0.0f;
#pragma unroll
    for (int k = 0; k < FIN; ++k) s = fmaf(xr[k], W[k * HID + c], s);
    h[tid] = s;
}

// ---------------- WMMA fp32 GEMMs ----------------
// One wave computes a 16-row tile: Y[16 x OUT] = X[16 x 32] @ W[32 x OUT]
// using V_WMMA_F32_16X16X4_F32, K-loop of 8.
// A (16x4): lane<16 -> {K=k0,k0+1}, lane>=16 -> {K=k0+2,k0+3} of row M=lane%16
// B (4x16): V0 = {K=k0 | K=k0+2}, V1 = {K=k0+1 | K=k0+3}, N = lane%16
// D (16x16): VGPR i -> M = i + 8*(lane/16), N = lane%16

__global__ void gemm32x32_wmma_kernel(const float* __restrict__ X, const float* __restrict__ W,
                                      float* __restrict__ Y, int n) {
    int gtid = blockIdx.x * blockDim.x + threadIdx.x;
    int wave = gtid >> 5;
    int lane = threadIdx.x & 31;
    int row0 = wave * 16;
    if (row0 >= n) return;
    int half = lane >> 4;   // 0 or 1
    int lr   = lane & 15;   // 0..15
    const float* xrow = X + (long long)(row0 + lr) * HID;
    v8f acc0 = {};
    v8f acc1 = {};
#pragma unroll
    for (int k0 = 0; k0 < 32; k0 += 4) {
        int kb = k0 + 2 * half;
        v2f a = *(const v2f*)(xrow + kb);
        v2f b0, b1;
        b0.x = W[(kb + 0) * 32 + lr];
        b0.y = W[(kb + 1) * 32 + lr];
        b1.x = W[(kb + 0) * 32 + lr + 16];
        b1.y = W[(kb + 1) * 32 + lr + 16];
        acc0 = __builtin_amdgcn_wmma_f32_16x16x4_f32(false, a, false, b0, (short)0, acc0, false, false);
        acc1 = __builtin_amdgcn_wmma_f32_16x16x4_f32(false, a, false, b1, (short)0, acc1, false, false);
    }
#pragma unroll
    for (int i = 0; i < 8; ++i) {
        int m = i + 8 * half;
        Y[(long long)(row0 + m) * 32 + lr]      = acc0[i];
        Y[(long long)(row0 + m) * 32 + lr + 16] = acc1[i];
    }
}

__global__ void gemm32x16_wmma_kernel(const float* __restrict__ X, const float* __restrict__ W,
                                      float* __restrict__ Y, int n) {
    int gtid = blockIdx.x * blockDim.x + threadIdx.x;
    int wave = gtid >> 5;
    int lane = threadIdx.x & 31;
    int row0 = wave * 16;
    if (row0 >= n) return;
    int half = lane >> 4;
    int lr   = lane & 15;
    const float* xrow = X + (long long)(row0 + lr) * HID;
    v8f acc = {};
#pragma unroll
    for (int k0 = 0; k0 < 32; k0 += 4) {
        int kb = k0 + 2 * half;
        v2f a = *(const v2f*)(xrow + kb);
        v2f b;
        b.x = W[(kb + 0) * 16 + lr];
        b.y = W[(kb + 1) * 16 + lr];
        acc = __builtin_amdgcn_wmma_f32_16x16x4_f32(false, a, false, b, (short)0, acc, false, false);
    }
#pragma unroll
    for (int i = 0; i < 8; ++i) {
        int m = i + 8 * half;
        Y[(long long)(row0 + m) * 16 + lr] = acc[i];
    }
}

// ---------------- scale by dinv[src] and zero the aggregation buffer ----------------

__global__ void scale_zero_kernel(const float* __restrict__ h, const float* __restrict__ dinv,
                                  float* __restrict__ hs, float* __restrict__ agg,
                                  int n, int F) {
    int tid = blockIdx.x * blockDim.x + threadIdx.x;
    if (tid >= n * F) return;
    int node = tid / F;
    hs[tid]  = h[tid] * dinv[node];
    agg[tid] = 0.0f;
}

// ---------------- edge scatter: agg[dst] += hs[src] * dinv[dst] ----------------
// F/4 threads per edge; per-thread float4 gather is coalesced across an edge's threads,
// atomics resolve in L2 (working set = N*F*4 bytes << 192 MB).

template <int F>
__global__ void scatter_kernel(const int* __restrict__ src, const int* __restrict__ dst,
                               const float* __restrict__ dinv, const float* __restrict__ hs,
                               float* __restrict__ agg, int E) {
    const int G = F / 4;
    long long tid = (long long)blockIdx.x * blockDim.x + threadIdx.x;
    if (tid >= (long long)E * G) return;
    int e = (int)(tid / G);
    int g = (int)(tid % G);
    int s = src[e];
    int d = dst[e];
    float dd = dinv[d];
    float4 v = *(const float4*)(hs + (long long)s * F + g * 4);
    float* ap = agg + (long long)d * F + g * 4;
    unsafeAtomicAdd(ap + 0, v.x * dd);
    unsafeAtomicAdd(ap + 1, v.y * dd);
    unsafeAtomicAdd(ap + 2, v.z * dd);
    unsafeAtomicAdd(ap + 3, v.w * dd);
}

// ---------------- finalize: + self-loop + bias (+ReLU) ----------------

__global__ void finalize_relu_kernel(const float* __restrict__ agg, const float* __restrict__ hs,
                                     const float* __restrict__ dinv, const float* __restrict__ bias,
                                     float* __restrict__ out, int n, int F) {
    int tid = blockIdx.x * blockDim.x + threadIdx.x;
    if (tid >= n * F) return;
    int node = tid / F;
    int c    = tid % F;
    float t = agg[tid] + hs[tid] * dinv[node] + bias[c];
    out[tid] = fmaxf(t, 0.0f);
}

// ---------------- final layer: finalize + log_softmax fused ----------------

__global__ void final_logsoftmax_kernel(const float* __restrict__ agg, const float* __restrict__ hs,
                                        const float* __restrict__ dinv, const float* __restrict__ bias,
                                        float* __restrict__ out, int n) {
    int node = blockIdx.x * blockDim.x + threadIdx.x;
    if (node >= n) return;
    float di = dinv[node];
    float v[NCLS];
    float mx = -3.4e38f;
#pragma unroll
    for (int c = 0; c < NCLS; ++c) {
        float t = agg[(long long)node * NCLS + c] + hs[(long long)node * NCLS + c] * di + bias[c];
        v[c] = t;
        mx = fmaxf(mx, t);
    }
    float se = 0.0f;
#pragma unroll
    for (int c = 0; c < NCLS; ++c) se += expf(v[c] - mx);
    float lse = logf(se) + mx;
#pragma unroll
    for (int c = 0; c < NCLS; ++c) out[(long long)node * NCLS + c] = v[c] - lse;
}

// ---------------- launch ----------------

static inline int cdiv(long long a, long long b) { return (int)((a + b - 1) / b); }

extern "C" void kernel_launch(void* const* d_in, const int* in_sizes, int n_in,
                              void* d_out, int out_size, void* d_ws, size_t ws_size,
                              hipStream_t stream) {
    const float* x  = (const float*)d_in[0];
    const int*   ei = (const int*)d_in[1];
    const float* W1 = (const float*)d_in[2];
    const float* b1 = (const float*)d_in[3];
    const float* W2 = (const float*)d_in[4];
    const float* b2 = (const float*)d_in[5];
    const float* W3 = (const float*)d_in[6];
    const float* b3 = (const float*)d_in[7];

    const int N = in_sizes[0] / FIN;
    const int E = in_sizes[1] / 2;
    const int* src = ei;       // edge_index[0]
    const int* dst = ei + E;   // edge_index[1]
    float* out = (float*)d_out;

    float* dinv = (float*)d_ws;                 // N
    float* bufA = dinv + N;                     // N*32
    float* bufB = bufA + (size_t)N * HID;       // N*32
    float* bufC = bufB + (size_t)N * HID;       // N*32

    const int T = 256;
    const long long NH  = (long long)N * HID;
    const long long NC  = (long long)N * NCLS;
    const int gemmThreads = ((N + 15) / 16) * 32;   // one wave per 16-row tile

    // degree (incl. self loop) -> dinv = rsqrt(deg)
    zero_kernel<<<cdiv(N, T), T, 0, stream>>>(dinv, N);
    deg_kernel<<<cdiv(E, T), T, 0, stream>>>(dst, dinv, E);
    dinv_kernel<<<cdiv(N, T), T, 0, stream>>>(dinv, N);

    // ---- layer 1: in=x, gemmOut=A, hs=B, agg=C, out=A ----
    gemm_in_kernel<<<cdiv(NH, T), T, 0, stream>>>(x, W1, bufA, N);
    scale_zero_kernel<<<cdiv(NH, T), T, 0, stream>>>(bufA, dinv, bufB, bufC, N, HID);
    scatter_kernel<HID><<<cdiv((long long)E * (HID / 4), T), T, 0, stream>>>(src, dst, dinv, bufB, bufC, E);
    finalize_relu_kernel<<<cdiv(NH, T), T, 0, stream>>>(bufC, bufB, dinv, b1, bufA, N, HID);

    // ---- layer 2: in=A, gemmOut=B, hs=C, agg=A, out=B ----
    gemm32x32_wmma_kernel<<<cdiv(gemmThreads, T), T, 0, stream>>>(bufA, W2, bufB, N);
    scale_zero_kernel<<<cdiv(NH, T), T, 0, stream>>>(bufB, dinv, bufC, bufA, N, HID);
    scatter_kernel<HID><<<cdiv((long long)E * (HID / 4), T), T, 0, stream>>>(src, dst, dinv, bufC, bufA, E);
    finalize_relu_kernel<<<cdiv(NH, T), T, 0, stream>>>(bufA, bufC, dinv, b2, bufB, N, HID);

    // ---- layer 3: in=B, gemmOut=C (N x 16), hs=A, agg=B, out=d_out ----
    gemm32x16_wmma_kernel<<<cdiv(gemmThreads, T), T, 0, stream>>>(bufB, W3, bufC, N);
    scale_zero_kernel<<<cdiv(NC, T), T, 0, stream>>>(bufC, dinv, bufA, bufB, N, NCLS);
    scatter_kernel<NCLS><<<cdiv((long long)E * (NCLS / 4), T), T, 0, stream>>>(src, dst, dinv, bufA, bufB, E);
    final_logsoftmax_kernel<<<cdiv(N, T), T, 0, stream>>>(bufB, bufA, dinv, b3, out, N);
}